// SGL_Encoder_12610023981257
// MI455X (gfx1250) — compile-verified
//
#include <hip/hip_runtime.h>
#include <hip/hip_bf16.h>
#include <stdint.h>

// Problem constants (match the reference).
#define USER_NUM 100000
#define ITEM_NUM 100000
#define N_NODES  (USER_NUM + ITEM_NUM)
#define EMB      64
#define N_EDGES  4000000
#define N_LAYERS 3

#define TILE_EDGES      2048      // edges staged per block via TDM (24 KB LDS)
#define SCATTER_BLOCK   256       // 8 waves (wave32)
#define WAVES_PER_BLOCK (SCATTER_BLOCK / 32)

typedef unsigned int v4u __attribute__((ext_vector_type(4)));
typedef int          v8i __attribute__((ext_vector_type(8)));
typedef int          v4i __attribute__((ext_vector_type(4)));

// ---------------------------------------------------------------------------
// Tensor Data Mover: 1D contiguous load of `tile_elems` 4-byte elements from
// `gaddr` into LDS byte-offset `lds_off`. D# layout per CDNA5 ISA §8:
//   group0: [1:0]=count=1, [63:32]=lds_addr, [120:64]=global_addr, [127:126]=2
//   group1: [17:16]=data_size(2 -> 4B), tensor_dim0, tensor_dim1=1,
//           tile_dim0=tile_elems, tile_dim1/2=0 (unused), dim0 stride
//   group2/3 (+ trailing group): zero (<=2D tensor)
// ---------------------------------------------------------------------------
__device__ __forceinline__ void tdm_load_1d_b32(unsigned lds_off,
                                                const void* gaddr,
                                                unsigned tile_elems,
                                                unsigned tensor_elems) {
  uint64_t ga = (uint64_t)(uintptr_t)gaddr;
  v4u g0;
  g0.x = 1u;                                   // count=1 (valid user descriptor)
  g0.y = lds_off;                              // lds_addr (bytes)
  g0.z = (unsigned)(ga & 0xFFFFFFFFu);         // global_addr[31:0]
  g0.w = (unsigned)((ga >> 32) & 0x1FFFFFFu)   // global_addr[56:32]
       | (2u << 30);                           // type = 2 ("image")
  v8i g1;
  g1[0] = (int)(2u << 16);                     // workgroup_mask=0, data_size=4B
  g1[1] = (int)((tensor_elems & 0xFFFFu) << 16);     // tensor_dim0[15:0]
  g1[2] = (int)((tensor_elems >> 16) | (1u << 16));  // tensor_dim0[31:16] | tensor_dim1=1
  g1[3] = (int)((tile_elems & 0xFFFFu) << 16);       // tile_dim0
  g1[4] = 0;                                   // tile_dim1=0, tile_dim2=0 (1D tile)
  g1[5] = (int)tensor_elems;                   // tensor_dim0_stride[31:0]
  g1[6] = 0;                                   // stride hi bits
  g1[7] = 0;
  v4i z4 = {0, 0, 0, 0};
  v8i z8 = {0, 0, 0, 0, 0, 0, 0, 0};
  __builtin_amdgcn_tensor_load_to_lds(g0, g1, z4, z4, z8, /*cpol=*/0);
}

// ---------------------------------------------------------------------------
// init: x0 = concat(user_emb, item_emb); acc(d_out) = x0; x1 = 0
// ---------------------------------------------------------------------------
__global__ void __launch_bounds__(256)
sgl_init_kernel(const float4* __restrict__ ue, const float4* __restrict__ ie,
                float4* __restrict__ x0, float4* __restrict__ acc,
                float4* __restrict__ x1, int n4_user, int n4_total) {
  int i = blockIdx.x * blockDim.x + threadIdx.x;
  if (i >= n4_total) return;
  float4 v = (i < n4_user) ? ue[i] : ie[i - n4_user];
  x0[i] = v;
  acc[i] = v;
  x1[i] = make_float4(0.f, 0.f, 0.f, 0.f);
}

// ---------------------------------------------------------------------------
// SpMM scatter. Edge records are DMA'd into LDS by the TDM (one wave issues,
// all consume). Compute: half-wave per edge -> 16 lanes x float4 covers the
// 64-float row with a single b128 gather (two independent 256B transactions
// per wave), 2x unrolled for ~4 outstanding gathers per wave. Scatter with
// no-return global_atomic_add_f32 (resolved at L2; all node matrices are
// L2-resident: 3 x 51.2 MB < 192 MB).
// ---------------------------------------------------------------------------
__global__ void __launch_bounds__(SCATTER_BLOCK)
sgl_spmm_scatter(const int* __restrict__ row, const int* __restrict__ col,
                 const float* __restrict__ val, const float* __restrict__ x,
                 float* __restrict__ y, int n_edges) {
  __shared__ int   sh_row[TILE_EDGES];
  __shared__ int   sh_col[TILE_EDGES];
  __shared__ float sh_val[TILE_EDGES];

  const int base = blockIdx.x * TILE_EDGES;
  int count = n_edges - base;
  if (count <= 0) return;
  if (count > TILE_EDGES) count = TILE_EDGES;

  if (threadIdx.x < 32) {  // one wave issues the three async DMAs
    const unsigned remaining = (unsigned)(n_edges - base);
    tdm_load_1d_b32((unsigned)(uintptr_t)sh_row, row + base, (unsigned)count, remaining);
    tdm_load_1d_b32((unsigned)(uintptr_t)sh_col, col + base, (unsigned)count, remaining);
    tdm_load_1d_b32((unsigned)(uintptr_t)sh_val, val + base, (unsigned)count, remaining);
    __builtin_amdgcn_s_wait_tensorcnt(0);
  }
  // Pad one neutral edge if the tile is odd-sized (disjoint from TDM's
  // write range [0, count), so no ordering hazard with the DMA).
  if (threadIdx.x == 33 && (count & 1)) {
    sh_row[count] = 0;
    sh_col[count] = 0;
    sh_val[count] = 0.f;
  }
  __syncthreads();

  const int lane = threadIdx.x & 31;
  const int wave = threadIdx.x >> 5;
  const int g    = lane >> 4;        // half-wave selects edge within a pair
  const int sub  = lane & 15;        // 16 lanes x float4 = 64 floats
  const int npairs = (count + 1) >> 1;

  for (int p = wave; p < npairs; p += 2 * WAVES_PER_BLOCK) {
    const int q = p + WAVES_PER_BLOCK;
    const bool dual = (q < npairs);

    const int   e0 = 2 * p + g;
    const int   r0 = sh_row[e0];
    const int   c0 = sh_col[e0];
    const float w0 = sh_val[e0];
    const float4 v0 = ((const float4*)(x + (size_t)c0 * EMB))[sub];

    int r1 = 0, c1 = 0;
    float w1 = 0.f;
    float4 v1 = make_float4(0.f, 0.f, 0.f, 0.f);
    if (dual) {
      const int e1 = 2 * q + g;
      r1 = sh_row[e1];
      c1 = sh_col[e1];
      w1 = sh_val[e1];
      v1 = ((const float4*)(x + (size_t)c1 * EMB))[sub];
    }

    float* d0 = y + (size_t)r0 * EMB + (sub << 2);
    unsafeAtomicAdd(d0 + 0, v0.x * w0);
    unsafeAtomicAdd(d0 + 1, v0.y * w0);
    unsafeAtomicAdd(d0 + 2, v0.z * w0);
    unsafeAtomicAdd(d0 + 3, v0.w * w0);
    if (dual) {
      float* d1 = y + (size_t)r1 * EMB + (sub << 2);
      unsafeAtomicAdd(d1 + 0, v1.x * w1);
      unsafeAtomicAdd(d1 + 1, v1.y * w1);
      unsafeAtomicAdd(d1 + 2, v1.z * w1);
      unsafeAtomicAdd(d1 + 3, v1.w * w1);
    }
  }
}

// ---------------------------------------------------------------------------
// commit: acc = (acc + x_next) * scale ; zero old x_cur so it becomes the
// next layer's (cleared) destination buffer.
// ---------------------------------------------------------------------------
__global__ void __launch_bounds__(256)
sgl_commit_kernel(float4* __restrict__ acc, const float4* __restrict__ xn,
                  float4* __restrict__ xz, float scale, int n4) {
  int i = blockIdx.x * blockDim.x + threadIdx.x;
  if (i >= n4) return;
  float4 a = acc[i];
  float4 b = xn[i];
  a.x = (a.x + b.x) * scale;
  a.y = (a.y + b.y) * scale;
  a.z = (a.z + b.z) * scale;
  a.w = (a.w + b.w) * scale;
  acc[i] = a;
  xz[i] = make_float4(0.f, 0.f, 0.f, 0.f);
}

// ---------------------------------------------------------------------------
// Inputs (setup_inputs order): user_emb f32[100000*64], item_emb f32[100000*64],
// adj_row i32[4M], adj_col i32[4M], adj_val f32[4M].
// Output: concat(user_out, item_out) f32[200000*64].
// Workspace: two ping-pong node matrices (2 * 51.2 MB).
// ---------------------------------------------------------------------------
extern "C" void kernel_launch(void* const* d_in, const int* in_sizes, int n_in,
                              void* d_out, int out_size, void* d_ws, size_t ws_size,
                              hipStream_t stream) {
  const float* ue  = (const float*)d_in[0];
  const float* ie  = (const float*)d_in[1];
  const int*   row = (const int*)d_in[2];
  const int*   col = (const int*)d_in[3];
  const float* val = (const float*)d_in[4];

  float* acc = (float*)d_out;
  float* x0  = (float*)d_ws;
  float* x1  = x0 + (size_t)N_NODES * EMB;

  const int n4_total = N_NODES * EMB / 4;   // 3,200,000 float4
  const int n4_user  = USER_NUM * EMB / 4;
  const int ew_grid  = (n4_total + 255) / 256;

  sgl_init_kernel<<<ew_grid, 256, 0, stream>>>(
      (const float4*)ue, (const float4*)ie,
      (float4*)x0, (float4*)acc, (float4*)x1, n4_user, n4_total);

  const int tiles = (N_EDGES + TILE_EDGES - 1) / TILE_EDGES;
  float* xc = x0;
  float* xn = x1;
  for (int l = 0; l < N_LAYERS; ++l) {
    sgl_spmm_scatter<<<tiles, SCATTER_BLOCK, 0, stream>>>(row, col, val, xc, xn, N_EDGES);
    const float scale = (l == N_LAYERS - 1) ? (1.0f / (N_LAYERS + 1)) : 1.0f;
    sgl_commit_kernel<<<ew_grid, 256, 0, stream>>>(
        (float4*)acc, (const float4*)xn, (float4*)xc, scale, n4_total);
    float* t = xc; xc = xn; xn = t;   // xn is now the freshly-zeroed buffer
  }
}